// MGN_NET_21096879358387
// MI455X (gfx1250) — compile-verified
//
#include <hip/hip_runtime.h>
#include <hip/hip_bf16.h>

// ---------------------------------------------------------------------------
// MGN-style NNConv x3 + pairwise L1 (CBT) for gfx1250 (MI455X).
// Heavy math (edge-MLP K=4 matmul) runs on V_WMMA_F32_16X16X4_F32, fp32 exact.
// ---------------------------------------------------------------------------

#define NODES 35
#define EDGES (NODES * (NODES - 1))   // 1190
#define ETILES ((EDGES + 15) / 16)    // 75
#define D1 512
#define D2 256
#define D3 128

typedef __attribute__((ext_vector_type(2))) float v2f;
typedef __attribute__((ext_vector_type(8))) float v8f;

// ---------------------------------------------------------------------------
// Fused per-edge message kernel:
//   msg[e, o] = sum_i x[src(e), i] * relu( edge_attr[e,:4] . W[:, i*OUT_C+o] + b[i*OUT_C+o] )
// One block = 16 edges; 8 waves, each wave owns a 16-wide column tile.
// Per i: one V_WMMA_F32_16X16X4_F32 computes c[16 edges, 16 cols] with the
// bias pre-loaded into the C accumulator (bias is constant per lane).
// i-loop unrolled x2 -> two independent WMMA chains in flight.
// src(e) derived analytically from the fixed fully-connected graph.
// ---------------------------------------------------------------------------
template <int IN_C, int OUT_C>
__global__ __launch_bounds__(256) void nnconv_msg_wmma(
    const float* __restrict__ x,      // [NODES, IN_C]
    const float* __restrict__ ea,     // [EDGES, 4]
    const float* __restrict__ mlp_w,  // [4, IN_C*OUT_C]
    const float* __restrict__ mlp_b,  // [IN_C*OUT_C]
    float* __restrict__ msg)          // [EDGES, OUT_C]
{
    __shared__ float xs[16 * IN_C];   // x[src(e)] staged for the 16 edges

    const int  tile      = blockIdx.x;
    const int  e0        = tile * 16;
    const bool full_tile = (e0 + 16 <= EDGES);
    const int  tid       = threadIdx.x;
    const int  wave      = tid >> 5;
    const int  lane      = tid & 31;
    const int  half      = lane >> 4;   // K-half for A/B operands
    const int  lm        = lane & 15;   // M for A, N for B/C/D

    // ---- stage source-node features for this edge tile into LDS ----
    for (int idx = tid; idx < 16 * IN_C; idx += 256) {
        const int m = idx / IN_C;
        const int i = idx - m * IN_C;
        const int e = e0 + m;
        float v = 0.0f;
        if (e < EDGES) {
            const int s = e / (NODES - 1);     // src node of edge e
            v = x[s * IN_C + i];
        }
        xs[idx] = v;
    }
    __syncthreads();

    // ---- A operand: edge_attr tile [16 edges x 4], per ISA 16x4 f32 layout ----
    v2f A;
    if (full_tile) {
        const int e = e0 + lm;
        A.x = ea[e * 4 + half * 2 + 0];
        A.y = ea[e * 4 + half * 2 + 1];
    } else {
        const int e = e0 + lm;
        if (e < EDGES) {
            A.x = ea[e * 4 + half * 2 + 0];
            A.y = ea[e * 4 + half * 2 + 1];
        } else {
            A.x = 0.0f; A.y = 0.0f;
        }
    }

    // ---- this wave's 16-column output tile ----
    const int o0 = blockIdx.y * 128 + wave * 16;
    const int oc = o0 + lm;           // this lane's output column

    float acc[8] = {0.f, 0.f, 0.f, 0.f, 0.f, 0.f, 0.f, 0.f};

#pragma unroll 2
    for (int i = 0; i < IN_C; ++i) {
        const int col = i * OUT_C + oc;

        // B operand: W[k, col] for k = half*2 + {0,1}  (4x16 f32 layout)
        const float* wp = mlp_w + (size_t)(half * 2) * (IN_C * OUT_C) + col;
        v2f B;
        B.x = wp[0];
        B.y = wp[(size_t)IN_C * OUT_C];

        // bias folded into the accumulator: constant per lane (same column)
        const float b = mlp_b[col];
        v8f C = {b, b, b, b, b, b, b, b};

        // D = A(16x4) x B(4x16) + C  -- full fp32, K=4 completes the reduction
        C = __builtin_amdgcn_wmma_f32_16x16x4_f32(
                false, A, false, B, (short)0, C, false, false);

        // relu, multiply by x[src, i], accumulate.  D layout: M = r + 8*half.
#pragma unroll
        for (int r = 0; r < 8; ++r) {
            float c = C[r];
            c = c > 0.0f ? c : 0.0f;
            acc[r] += xs[(r + half * 8) * IN_C + i] * c;
        }
    }

    // ---- store: straight-line fast path for the 74/75 full tiles ----
    if (full_tile) {
#pragma unroll
        for (int r = 0; r < 8; ++r)
            msg[(size_t)(e0 + r + half * 8) * OUT_C + oc] = acc[r];
    } else {
#pragma unroll
        for (int r = 0; r < 8; ++r) {
            const int e = e0 + r + half * 8;
            if (e < EDGES) msg[(size_t)e * OUT_C + oc] = acc[r];
        }
    }
}

// ---------------------------------------------------------------------------
// Aggregation: scatter-mean over incoming edges (analytic full-graph inverse
// map), plus root matmul, bias, relu.
//   h[n,d] = relu( mean_{m!=n} msg[e(m->n), d] + sum_i x[n,i]*root[i,d] + bias[d] )
// ---------------------------------------------------------------------------
template <int IN_C, int OUT_C>
__global__ __launch_bounds__(256) void nnconv_agg(
    const float* __restrict__ msg,   // [EDGES, OUT_C]
    const float* __restrict__ xin,   // [NODES, IN_C]
    const float* __restrict__ root,  // [IN_C, OUT_C]
    const float* __restrict__ bias,  // [OUT_C]
    float* __restrict__ h)           // [NODES, OUT_C]
{
    const int n = blockIdx.x;
    for (int d = threadIdx.x; d < OUT_C; d += blockDim.x) {
        float s = 0.0f;
        for (int m = 0; m < NODES; ++m) {
            if (m == n) continue;
            const int e = m * (NODES - 1) + n - (n > m ? 1 : 0);
            s += msg[(size_t)e * OUT_C + d];
        }
        s *= (1.0f / (float)(NODES - 1));

        float rt = 0.0f;
        for (int i = 0; i < IN_C; ++i)
            rt += xin[n * IN_C + i] * root[i * OUT_C + d];

        const float v = s + rt + bias[d];
        h[n * OUT_C + d] = v > 0.0f ? v : 0.0f;
    }
}

// ---------------------------------------------------------------------------
// cbt[a, i] = sum_f |h[i, f] - h[a, f]|
// ---------------------------------------------------------------------------
__global__ __launch_bounds__(256) void cbt_kernel(
    const float* __restrict__ h,     // [NODES, D3]
    float* __restrict__ out)         // [NODES, NODES]
{
    const int p = blockIdx.x * blockDim.x + threadIdx.x;
    if (p >= NODES * NODES) return;
    const int a = p / NODES;
    const int i = p - a * NODES;
    float s = 0.0f;
    for (int f = 0; f < D3; ++f)
        s += fabsf(h[i * D3 + f] - h[a * D3 + f]);
    out[p] = s;
}

// ---------------------------------------------------------------------------
extern "C" void kernel_launch(void* const* d_in, const int* in_sizes, int n_in,
                              void* d_out, int out_size, void* d_ws, size_t ws_size,
                              hipStream_t stream) {
    const float* x   = (const float*)d_in[0];   // [35, 1]
    const float* ea  = (const float*)d_in[1];   // [1190, 4]
    // d_in[2] = edge_index (fixed full graph; derived analytically instead)
    const float* w1  = (const float*)d_in[3];
    const float* b1  = (const float*)d_in[4];
    const float* r1  = (const float*)d_in[5];
    const float* bb1 = (const float*)d_in[6];
    const float* w2  = (const float*)d_in[7];
    const float* b2  = (const float*)d_in[8];
    const float* r2  = (const float*)d_in[9];
    const float* bb2 = (const float*)d_in[10];
    const float* w3  = (const float*)d_in[11];
    const float* b3  = (const float*)d_in[12];
    const float* r3  = (const float*)d_in[13];
    const float* bb3 = (const float*)d_in[14];

    float* ws  = (float*)d_ws;
    float* msg = ws;                          // max EDGES*D1 = 609,280 floats
    float* h1  = msg + (size_t)EDGES * D1;    // [35, 512]
    float* h2  = h1 + NODES * D1;             // [35, 256]
    float* h3  = h2 + NODES * D2;             // [35, 128]

    // ---- layer 1 (in=1, out=512) ----
    nnconv_msg_wmma<1, D1><<<dim3(ETILES, D1 / 128), 256, 0, stream>>>(x, ea, w1, b1, msg);
    nnconv_agg<1, D1><<<NODES, 256, 0, stream>>>(msg, x, r1, bb1, h1);

    // ---- layer 2 (in=512, out=256) ----
    nnconv_msg_wmma<D1, D2><<<dim3(ETILES, D2 / 128), 256, 0, stream>>>(h1, ea, w2, b2, msg);
    nnconv_agg<D1, D2><<<NODES, 256, 0, stream>>>(msg, h1, r2, bb2, h2);

    // ---- layer 3 (in=256, out=128) ----
    nnconv_msg_wmma<D2, D3><<<dim3(ETILES, D3 / 128), 256, 0, stream>>>(h2, ea, w3, b3, msg);
    nnconv_agg<D2, D3><<<NODES, 256, 0, stream>>>(msg, h2, r3, bb3, h3);

    // ---- pairwise L1 ----
    cbt_kernel<<<(NODES * NODES + 255) / 256, 256, 0, stream>>>(h3, (float*)d_out);
}